// StochasticPool2d_1580547974630
// MI455X (gfx1250) — compile-verified
//
#include <hip/hip_runtime.h>
#include <stdint.h>

#define B_  32
#define C_  64
#define H_  224
#define W_  224
#define NH  112
#define NW  112
#define RPB 4            // sampled output rows per block

typedef __attribute__((ext_vector_type(4))) unsigned int u32x4;
typedef __attribute__((ext_vector_type(8))) int          i32x8;
typedef __attribute__((ext_vector_type(4))) int          i32x4;

// One block per (b, c, oh-group of 4). Wave 0 issues 4 pipelined TDM loads
// (rows h_r, h_r+1 of x[b,c] -> LDS), single s_wait_tensorcnt, then all
// threads gather stochastic column pairs from LDS and take 2x2 maxes.
__global__ __launch_bounds__(128) void stochpool_tdm(const float* __restrict__ x,
                                                     const int*   __restrict__ h_off,
                                                     const int*   __restrict__ w_off,
                                                     float*       __restrict__ out) {
    __shared__ float tile[RPB][2 * W_];     // 4 x 1792 B

    const int g  = blockIdx.x;              // 0..27  (oh group)
    const int c  = blockIdx.y;              // 0..63
    const int b  = blockIdx.z;              // 0..31
    const int tx = threadIdx.x;

    int hrow[RPB];                          // block-uniform -> scalar loads
#pragma unroll
    for (int r = 0; r < RPB; ++r) {
        const int oh = RPB * g + r;
        hrow[r] = 2 * oh + h_off[oh];       // 0..223
    }

    // Pre-zero LDS row 1 of any tile whose second pool row is the zero-pad
    // row (h == 223): the TDM will load only one row there (tile_dim1 = 1),
    // so this region is disjoint from all TDM writes -> no extra barrier.
#pragma unroll
    for (int r = 0; r < RPB; ++r) {
        if (hrow[r] == H_ - 1) {            // uniform condition
            if (tx < NW) {
                tile[r][W_ + tx]      = 0.0f;
                tile[r][W_ + NW + tx] = 0.0f;
            }
        }
    }

    const float* plane = x + ((size_t)b * C_ + c) * H_ * W_;

#if __has_builtin(__builtin_amdgcn_tensor_load_to_lds)
    if (tx < 32) {                          // wave 0 issues all DMAs
#pragma unroll
        for (int r = 0; r < RPB; ++r) {
            const int      h         = hrow[r];
            const uint64_t ga        = (uint64_t)(uintptr_t)(plane + (size_t)h * W_);
            const uint32_t lds       = (uint32_t)(uintptr_t)&tile[r][0]; // low 32b = LDS offset
            const uint32_t rows_rem  = (uint32_t)(H_ - h);
            const uint32_t tile_rows = rows_rem < 2u ? rows_rem : 2u;

            // ---- D# group 0: count=1 | lds_addr | global_addr | type=2 ----
            u32x4 g0;
            g0.x = 1u;                                   // count=1, user descriptor
            g0.y = lds;                                  // lds_addr[31:0]
            g0.z = (uint32_t)ga;                         // global_addr[31:0]
            g0.w = (uint32_t)(ga >> 32) | 0x80000000u;   // global_addr[56:32], type=2

            // ---- D# group 1 ----
            i32x8 g1;
            g1[0] = 0x00020000;                 // wg_mask=0, data_size=2 (4 B), no flags
            g1[1] = (int)((uint32_t)W_ << 16);  // abar_addr=0 | tensor_dim0[15:0]=224
            g1[2] = (int)(rows_rem << 16);      // dim0[31:16]=0 | tensor_dim1[15:0]=224-h
            g1[3] = (int)((uint32_t)W_ << 16);  // dim1[31:16]=0 | tile_dim0=224
            g1[4] = (int)tile_rows;             // tile_dim1 (1|2) | tile_dim2=0
            g1[5] = W_;                         // tensor_dim0_stride[31:0]=224
            g1[6] = 0;                          // stride0[47:32] | stride1[15:0]
            g1[7] = 0;                          // stride1[47:16]

            i32x4 gz = {0, 0, 0, 0};            // groups 2/3 unused (2D tensor)
#if __clang_major__ >= 23
            i32x8 g4 = {0, 0, 0, 0, 0, 0, 0, 0};
            __builtin_amdgcn_tensor_load_to_lds(g0, g1, gz, gz, g4, 0);
#else
            __builtin_amdgcn_tensor_load_to_lds(g0, g1, gz, gz, 0);
#endif
        }
        __builtin_amdgcn_s_wait_tensorcnt(0);   // all 4 transfers complete
    }
    __syncthreads();
#else
    // Fallback: cooperative coalesced staging (no TDM builtin available)
    for (int r = 0; r < RPB; ++r) {
        const int h = hrow[r];
        for (int i = tx; i < 2 * W_; i += blockDim.x) {
            const int rr = i / W_, cc = i % W_;
            tile[r][i] = (h + rr < H_) ? plane[(size_t)(h + rr) * W_ + cc] : 0.0f;
        }
    }
    __syncthreads();
#endif

    if (tx < NW) {
        const int  w   = 2 * tx + w_off[tx];    // 0..223, reused for all 4 rows
        const bool inb = (w + 1 < W_);          // right-edge zero padding
        float* orow = out + (((size_t)b * C_ + c) * NH + (size_t)RPB * g) * NW + tx;
#pragma unroll
        for (int r = 0; r < RPB; ++r) {
            const float v00 = tile[r][w];
            const float v10 = tile[r][W_ + w];
            const float v01 = inb ? tile[r][w + 1]      : 0.0f;
            const float v11 = inb ? tile[r][W_ + w + 1] : 0.0f;
            orow[(size_t)r * NW] = fmaxf(fmaxf(v00, v01), fmaxf(v10, v11));
        }
    }
}

extern "C" void kernel_launch(void* const* d_in, const int* in_sizes, int n_in,
                              void* d_out, int out_size, void* d_ws, size_t ws_size,
                              hipStream_t stream) {
    const float* x     = (const float*)d_in[0];
    const int*   h_off = (const int*)d_in[1];
    const int*   w_off = (const int*)d_in[2];
    float*       out   = (float*)d_out;

    dim3 grid(NH / RPB, C_, B_);  // (oh-group, c, b) = 28 x 64 x 32 blocks
    dim3 block(128);              // 4 wave32 per block
    stochpool_tdm<<<grid, block, 0, stream>>>(x, h_off, w_off, out);
}